// DynamicAttentionLayer_45732811767851
// MI455X (gfx1250) — compile-verified
//
#include <hip/hip_runtime.h>
#include <cstddef>
#include <cstdint>

// Problem constants (from reference)
#define NN 16384
#define MM 8192
#define EE 524288
#define DD 128      // distribution dim
#define IND 128     // feature in-dim
#define OUTD 64     // feature out-dim
#define LRELU_A 0.2f
#define CAP 4096    // per-row nonzero compaction capacity

typedef __attribute__((ext_vector_type(2))) float v2f;
typedef __attribute__((ext_vector_type(8))) float v8f;
typedef __attribute__((ext_vector_type(4))) int   v4i;

#define AS1 __attribute__((address_space(1)))
#define AS3 __attribute__((address_space(3)))

// CDNA5 async global->LDS path (ASYNCcnt-tracked), guarded so we compile on
// toolchains that lack the builtins.
#if defined(__has_builtin)
#if __has_builtin(__builtin_amdgcn_global_load_async_to_lds_b128) && \
    __has_builtin(__builtin_amdgcn_s_wait_asynccnt)
#define HAVE_ASYNC_LDS 1
#endif
#endif
#ifndef HAVE_ASYNC_LDS
#define HAVE_ASYNC_LDS 0
#endif

// Order-preserving float<->uint mapping for atomicMax on floats.
__device__ __forceinline__ unsigned f32_orderable(float f) {
  unsigned u = __float_as_uint(f);
  return (u & 0x80000000u) ? ~u : (u | 0x80000000u);
}
__device__ __forceinline__ float f32_unorder(unsigned u) {
  return (u & 0x80000000u) ? __uint_as_float(u & 0x7fffffffu)
                           : __uint_as_float(~u);
}

// K1: fold projection vectors through weight matrices:
// wa1 = W@a1, wa2 = W@a2, dwa1 = dW@da1, dwa2 = dW@da2 (each 128 floats);
// also zero the shared zero-vector used as the B-fragment filler.
__global__ void k_combine_params(const float* __restrict__ W, const float* __restrict__ ap,
                                 const float* __restrict__ dW, const float* __restrict__ dap,
                                 float* __restrict__ wa1, float* __restrict__ wa2,
                                 float* __restrict__ dwa1, float* __restrict__ dwa2,
                                 float* __restrict__ zvec) {
  int k = threadIdx.x;
  if (k >= IND) return;
  float s1 = 0.f, s2 = 0.f, s3 = 0.f, s4 = 0.f;
  for (int j = 0; j < OUTD; ++j) {
    float w = W[k * OUTD + j], dw = dW[k * OUTD + j];
    s1 += w * ap[j];
    s2 += w * ap[OUTD + j];
    s3 += dw * dap[j];
    s4 += dw * dap[OUTD + j];
  }
  wa1[k] = s1; wa2[k] = s2; dwa1[k] = s3; dwa2[k] = s4;
  zvec[k] = 0.f;
}

// K2: batched row-dot via V_WMMA_F32_16X16X4_F32.
// One wave handles a 16-row tile of A (rows x 128). B (4x16) column 0 = v0,
// column 1 = v1. Accumulated C column 0 = A@v0, column 1 = A@v1.
// Branchless B fragment: each lane selects its column source pointer ONCE
// (v0 / v1 / shared zero vector), so the hot loop is loads + wmma only.
// A 16x4 f32 frag: lanes 0-15 hold K={kk,kk+1}, lanes 16-31 hold K={kk+2,kk+3}.
// B 4x16 f32 frag: VGPR0 -> row kk+2*half, VGPR1 -> row kk+1+2*half, col = lane&15.
// C 16x16 f32: VGPR r -> M=r (lanes 0-15, N=lane) / M=8+r (lanes 16-31).
__global__ __launch_bounds__(256) void k_rowdot(const float* __restrict__ A,
                                                const float* __restrict__ v0,
                                                const float* __restrict__ v1,
                                                const float* __restrict__ zvec,
                                                float* __restrict__ out0,
                                                float* __restrict__ out1) {
  const int wave = threadIdx.x >> 5;
  const int lane = threadIdx.x & 31;
  const int tile = blockIdx.x * 8 + wave;
  const int base = tile * 16;
  const int half = lane >> 4;
  const int lr   = lane & 15;
  const float* arow = A + (size_t)(base + lr) * IND;
  const float* bptr = (lr == 0) ? v0 : ((lr == 1) ? v1 : zvec);

  v8f c = {};
#pragma unroll
  for (int kk = 0; kk < IND; kk += 4) {
    const int k0 = kk + 2 * half;
    float2 af = *(const float2*)(arow + k0);   // 8B aligned (k0 even)
    float2 bf = *(const float2*)(bptr + k0);
    v2f a; a.x = af.x; a.y = af.y;
    v2f b; b.x = bf.x; b.y = bf.y;
    c = __builtin_amdgcn_wmma_f32_16x16x4_f32(false, a, false, b, (short)0, c,
                                              false, false);
  }
  if (lr == 0) {
#pragma unroll
    for (int r = 0; r < 8; ++r) out0[base + 8 * half + r] = c[r];
  } else if (lr == 1 && out1) {
#pragma unroll
    for (int r = 0; r < 8; ++r) out1[base + 8 * half + r] = c[r];
  }
}

// K3: alpha = sigmoid(t1+t2); also init per-row max accumulators.
__global__ void k_alpha_init(const float* __restrict__ t1, const float* __restrict__ t2,
                             float* __restrict__ alpha_out, unsigned* __restrict__ rowmax_u) {
  int i = blockIdx.x * blockDim.x + threadIdx.x;
  if (i < NN) {
    float de = t1[i] + t2[i];
    alpha_out[i] = 1.0f / (1.0f + expf(-de));
    rowmax_u[i] = 0u;  // below every orderable-encoded real float
  }
}

// K4: zero-fill the dense attention output region (float4 stores).
__global__ void k_fill_zero(float4* __restrict__ p, size_t n4) {
  size_t i = (size_t)blockIdx.x * blockDim.x + threadIdx.x;
  if (i < n4) p[i] = make_float4(0.f, 0.f, 0.f, 0.f);
}

// K5: per-edge row-max (idempotent atomic max -> deterministic).
__global__ void k_edge_rowmax(const int* __restrict__ adj, const float* __restrict__ sa,
                              const float* __restrict__ sb, unsigned* __restrict__ rowmax_u) {
  int k = blockIdx.x * blockDim.x + threadIdx.x;
  if (k >= EE) return;
  int s = adj[k], d = adj[EE + k];
  float e = sa[s] + sb[d];
  e = (e > 0.f) ? e : LRELU_A * e;
  atomicMax(&rowmax_u[s], f32_orderable(e));
}

// K6: scatter unnormalized exp weights into the dense matrix.
// Duplicate edges write the identical value -> matches JAX .at[].set dedupe.
__global__ void k_edge_scatter(const int* __restrict__ adj, const float* __restrict__ sa,
                               const float* __restrict__ sb,
                               const unsigned* __restrict__ rowmax_u,
                               float* __restrict__ att) {
  int k = blockIdx.x * blockDim.x + threadIdx.x;
  if (k >= EE) return;
  int s = adj[k], d = adj[EE + k];
  float e = sa[s] + sb[d];
  e = (e > 0.f) ? e : LRELU_A * e;
  float rm = f32_unorder(rowmax_u[s]);
  att[(size_t)s * MM + d] = expf(e - rm);
}

// K7: one block per row. Stage 32KB row in LDS (async global->LDS on CDNA5),
// reduce sum, normalize+write attention, deterministically compact nonzeros,
// sparse-accumulate y = att@b_dis, blend with alpha/da_dis, tanh, L1-normalize.
__global__ __launch_bounds__(256) void k_row_softmax_spmm(
    const float* __restrict__ b_dis, const float* __restrict__ da_dis,
    const float* __restrict__ alpha_v, float* __restrict__ att,
    float* __restrict__ out_dis) {
  const int i   = blockIdx.x;
  const int tid = threadIdx.x;
  float* attrow = att + (size_t)i * MM;

  __shared__ float          srow[MM];   // 32 KB
  __shared__ float          wred[8];
  __shared__ int            scnt[256];
  __shared__ unsigned short sidx[CAP];  // 8 KB
  __shared__ float          swv[CAP];   // 16 KB
  __shared__ float          ycomb[256];
  __shared__ float          l1p[4];

  const int jb = tid * 32;
  float psum = 0.f;
#if HAVE_ASYNC_LDS
  // Fire-and-forget async copies straight into LDS (no VGPR staging),
  // tracked with ASYNCcnt; then sum from LDS.
#pragma unroll
  for (int q = 0; q < 8; ++q)
    __builtin_amdgcn_global_load_async_to_lds_b128(
        (AS1 v4i*)(attrow + jb + 4 * q),
        (AS3 v4i*)(srow + jb + 4 * q), 0, 0);
  __builtin_amdgcn_s_wait_asynccnt(0);
  __syncthreads();
#pragma unroll
  for (int q = 0; q < 8; ++q) {
    float4 v = *(const float4*)(srow + jb + 4 * q);
    psum += (v.x + v.y) + (v.z + v.w);
  }
#else
#pragma unroll
  for (int q = 0; q < 8; ++q) {
    float4 v = *(const float4*)(attrow + jb + 4 * q);
    *(float4*)(srow + jb + 4 * q) = v;
    psum += (v.x + v.y) + (v.z + v.w);
  }
#endif
#pragma unroll
  for (int off = 16; off > 0; off >>= 1) psum += __shfl_down(psum, off);
  if ((tid & 31) == 0) wred[tid >> 5] = psum;
  __syncthreads();
  float rowsum = 0.f;
#pragma unroll
  for (int q = 0; q < 8; ++q) rowsum += wred[q];   // block-uniform

  const int d  = tid & 127;
  const int hh = tid >> 7;
  float y = 0.f;

  if (rowsum > 0.f) {
    const float inv = 1.0f / rowsum;
    int cnt = 0;
    for (int q = 0; q < 32; ++q) cnt += (srow[jb + q] != 0.f) ? 1 : 0;
    scnt[tid] = cnt;
    __syncthreads();
    // Hillis-Steele inclusive scan (deterministic compaction order)
    for (int off = 1; off < 256; off <<= 1) {
      int vself = scnt[tid];
      int vprev = (tid >= off) ? scnt[tid - off] : 0;
      __syncthreads();
      scnt[tid] = vself + vprev;
      __syncthreads();
    }
    const int total = scnt[255];
    const int wbase = scnt[tid] - cnt;

    // normalized attention write-back (vectorized)
#pragma unroll
    for (int q = 0; q < 8; ++q) {
      float4 v = *(const float4*)(srow + jb + 4 * q);
      v.x *= inv; v.y *= inv; v.z *= inv; v.w *= inv;
      *(float4*)(attrow + jb + 4 * q) = v;
    }
    if (total <= CAP) {
      int p = wbase;
      for (int q = 0; q < 32; ++q) {
        float w = srow[jb + q];
        if (w != 0.f) { sidx[p] = (unsigned short)(jb + q); swv[p] = w * inv; ++p; }
      }
    }
    __syncthreads();
    if (total <= CAP) {
      // ~32 nonzeros expected; two j-interleaved halves combined in fixed order
      for (int k = hh; k < total; k += 2)
        y += swv[k] * b_dis[(size_t)sidx[k] * DD + d];
      ycomb[tid] = y;
      __syncthreads();
      y = (tid < 128) ? (ycomb[tid] + ycomb[tid + 128]) : 0.f;
    } else {
      if (tid < 128) {
        for (int j = 0; j < MM; ++j) {
          float w = srow[j];
          if (w != 0.f) y += (w * inv) * b_dis[(size_t)j * DD + d];
        }
      }
    }
  } else {
    // empty row: softmax of all-equal NEG_FILL logits -> uniform 1/M
    const float u = 1.0f / (float)MM;
#pragma unroll
    for (int q = 0; q < 8; ++q)
      *(float4*)(attrow + jb + 4 * q) = make_float4(u, u, u, u);
    if (tid < 128) {
      float acc = 0.f;
      for (int j = 0; j < MM; ++j) acc += b_dis[(size_t)j * DD + d];
      y = acc * u;
    }
  }

  float t = 0.f;
  if (tid < 128) {
    float al = alpha_v[i];
    float v  = (1.0f - al) * y + al * da_dis[(size_t)i * DD + d];
    t = tanhf(v);
  }
  float aab = fabsf(t);
#pragma unroll
  for (int off = 16; off > 0; off >>= 1) aab += __shfl_down(aab, off);
  if ((tid & 31) == 0 && tid < 128) l1p[tid >> 5] = aab;
  __syncthreads();
  if (tid < 128) {
    float l1 = (l1p[0] + l1p[1]) + (l1p[2] + l1p[3]);
    out_dis[(size_t)i * DD + d] = t / fmaxf(l1, 1e-12f);
  }
}

extern "C" void kernel_launch(void* const* d_in, const int* in_sizes, int n_in,
                              void* d_out, int out_size, void* d_ws, size_t ws_size,
                              hipStream_t stream) {
  const float* a_feats  = (const float*)d_in[0];
  const float* b_feats  = (const float*)d_in[1];
  const float* da_feats = (const float*)d_in[2];
  const float* b_dis    = (const float*)d_in[3];
  const float* da_dis   = (const float*)d_in[4];
  const int*   adj      = (const int*)d_in[5];
  const float* W        = (const float*)d_in[6];
  const float* a_param  = (const float*)d_in[7];
  const float* dW       = (const float*)d_in[8];
  const float* da_param = (const float*)d_in[9];

  float* out       = (float*)d_out;
  float* out_dis   = out;                               // N*128
  float* out_att   = out + (size_t)NN * DD;             // N*M
  float* out_alpha = out_att + (size_t)NN * MM;         // N

  float* ws   = (float*)d_ws;
  float* wa1  = ws;
  float* wa2  = ws + 128;
  float* dwa1 = ws + 256;
  float* dwa2 = ws + 384;
  float* zvec = ws + 512;          // 128 zeros (B-fragment filler)
  float* sa   = ws + 640;          // N
  float* sb   = sa + NN;           // M
  float* t1   = sb + MM;           // N
  float* t2   = t1 + NN;           // N
  unsigned* rowmax_u = (unsigned*)(t2 + NN);  // N

  // K1: tiny param folds (+ zero vector init)
  k_combine_params<<<1, 128, 0, stream>>>(W, a_param, dW, da_param,
                                          wa1, wa2, dwa1, dwa2, zvec);

  // K2: WMMA batched row-dots (each wave = 16 rows)
  k_rowdot<<<NN / 128, 256, 0, stream>>>(a_feats,  wa1,  dwa1, zvec, sa, t1);
  k_rowdot<<<MM / 128, 256, 0, stream>>>(b_feats,  wa2,  zvec, zvec, sb, nullptr);
  k_rowdot<<<NN / 128, 256, 0, stream>>>(da_feats, dwa2, zvec, zvec, t2, nullptr);

  // K3: alpha + rowmax init
  k_alpha_init<<<NN / 256, 256, 0, stream>>>(t1, t2, out_alpha, rowmax_u);

  // K4: zero dense attention region
  size_t n4 = (size_t)NN * MM / 4;
  k_fill_zero<<<(unsigned)(n4 / 256), 256, 0, stream>>>((float4*)out_att, n4);

  // K5/K6: edge passes
  k_edge_rowmax <<<EE / 256, 256, 0, stream>>>(adj, sa, sb, rowmax_u);
  k_edge_scatter<<<EE / 256, 256, 0, stream>>>(adj, sa, sb, rowmax_u, out_att);

  // K7: fused softmax-normalize + sparse SpMM + blend/tanh/L1 per row
  k_row_softmax_spmm<<<NN, 256, 0, stream>>>(b_dis, da_dis, out_alpha, out_att, out_dis);
}